// InverseWaveletLayer_21638045237628
// MI455X (gfx1250) — compile-verified
//
#include <hip/hip_runtime.h>

// Fused 4-level inverse DWT (8-tap reconstruction filters) for MI455X / gfx1250.
//
// Design (compile-only reasoning):
//  * Memory-bound (~134 MB HBM traffic at 23.3 TB/s ~ 6 us): fuse all four
//    levels. One workgroup per (b,c) row; the 2054/4102/8196 f32 intermediates
//    live in LDS (48 KB of the 320 KB WGP LDS); inputs read once, output
//    written once.
//  * Per-level stencil  y[2m+j] = sum_t lo[6+j-2t] a[m+t] + hi[6+j-2t] d[m+t]
//    computed with V_WMMA_F32_16X16X4_F32 (true fp32): M = 16 independent
//    segments (m0 = base+8M), N = 16 outputs/segment, K accumulates taps
//    i = 0..10 in three K=4 chunks x two filters -> 6 chained WMMAs per
//    256-output tile.  Hot loop: 6x global_load_b64 -> 6x chained wmma ->
//    4x ds_store_2addr_b32 (or global stores for the last level).
//  * Tile loop counter is readfirstlane'd into an SGPR so loop control is
//    scalar (s_cmp/s_cbranch) and EXEC is trivially all-ones across WMMAs.
//  * Full tiles take an unconditional path (only tap i=11 is clamped; its B
//    column is all-zero so the clamped value is inert). The 6/6/4/0-element
//    tails are a per-thread scalar stencil on <8 threads.

typedef __attribute__((ext_vector_type(2))) float v2f;
typedef __attribute__((ext_vector_type(8))) float v8f;

#define ROWS      1024      // B*C = 64*16
#define N_IN0     1030      // approx / d0 length
#define N_IN1     2054      // d1 length == a1 length
#define N_IN2     4101      // d2 length (a2 is 4102, trimmed)
#define N_IN3     8195      // d3 length (a3 is 8196, trimmed)
#define N_A2BUF   4102
#define N_A3BUF   8196
#define N_OUT     16384
#define NWAVES    8         // 256-thread block, wave32

// Banded reconstruction-filter weight W[i][N]:
//   N = 2u + j ; contribution of input x[m0+i] to output y[2(m0+u)+j]
//   = filt[6+j-2t] with t = i-u, valid for 0 <= t <= 3, else 0.
__device__ __forceinline__ float idwt_wgt(const float* __restrict__ f, int i, int N) {
  const int u = N >> 1;
  const int j = N & 1;
  const int t = i - u;
  return (t >= 0 && t <= 3) ? f[6 + j - 2 * t] : 0.0f;
}

// One IDWT level: inputs a[0..n-1], d[0..n-1] -> out[0..2n-7].
// a/out may point into LDS (generic address space) or global memory.
// Ba[c]/Bh[c]: this lane's B-matrix registers for K-chunk c
// (B 4x16 f32 layout: VGPR v, lane-half h -> row K = v + 2h, col N = lane&15).
template <int n>
__device__ __forceinline__ void idwt_level(
    const float* __restrict__ a, const float* __restrict__ d,
    float* __restrict__ out,
    int tid, int wave_s, int h, int Ml,
    const v2f* __restrict__ Ba, const v2f* __restrict__ Bh,
    const float* __restrict__ rec_lo, const float* __restrict__ rec_hi)
{
  constexpr int out_len = 2 * n - 6;
  constexpr int nFull   = out_len >> 8;   // full 256-output tiles
  constexpr int tail    = out_len & 255;  // 6 / 6 / 4 / 0

  // wave_s is SGPR-resident (readfirstlane) -> scalar loop control.
  for (int t = wave_s; t < nFull; t += NWAVES) {
    const int base_m = t << 7;            // 128 input positions per tile
    const int m0     = base_m + 8 * Ml;   // this lane's segment (A-row M = Ml)
    v8f acc = {0.f, 0.f, 0.f, 0.f, 0.f, 0.f, 0.f, 0.f};
#pragma unroll
    for (int c = 0; c < 3; ++c) {
      // A 16x4 f32 layout: lane-half h, VGPR v -> K = v + 2h  => this lane
      // loads inputs at tap index i = 4c + 2h (+0, +1).
      const int idx = m0 + 4 * c + 2 * h;
      int i1 = idx + 1;
      if (c == 2) i1 = (i1 <= n - 1) ? i1 : (n - 1);  // tap i=11: zero-weight col
      v2f Aa, Ad;
      Aa.x = a[idx];  Aa.y = a[i1];
      Ad.x = d[idx];  Ad.y = d[i1];
      // D = A x B + C  (16x16x4, fp32)
      acc = __builtin_amdgcn_wmma_f32_16x16x4_f32(false, Aa, false, Ba[c],
                                                  (short)0, acc, false, false);
      acc = __builtin_amdgcn_wmma_f32_16x16x4_f32(false, Ad, false, Bh[c],
                                                  (short)0, acc, false, false);
    }
    // D 16x16 f32 layout: lane L, VGPR v -> M = v + 8h, N = Ml.
    // Output position = 2*base_m + 16*M + N ; all 256 valid in a full tile.
    const int obase = 2 * base_m + 128 * h + Ml;
#pragma unroll
    for (int v = 0; v < 8; ++v) out[obase + 16 * v] = acc[v];
  }

  // Tail outputs (< 8 per level): per-thread scalar stencil; filter taps
  // fetched directly (runtime index) to avoid a cndmask select tree.
  if constexpr (tail > 0) {
    if (tid < tail) {
      const int o = nFull * 256 + tid;
      const int m = o >> 1;
      const int j = o & 1;
      float acc = 0.f;
#pragma unroll
      for (int t = 0; t < 4; ++t) {
        acc += rec_lo[6 + j - 2 * t] * a[m + t]
             + rec_hi[6 + j - 2 * t] * d[m + t];
      }
      out[o] = acc;
    }
  }
}

__global__ __launch_bounds__(256)
void idwt4_fused_kernel(const float* __restrict__ approx,
                        const float* __restrict__ d0,
                        const float* __restrict__ d1,
                        const float* __restrict__ d2,
                        const float* __restrict__ d3,
                        const float* __restrict__ rec_lo,
                        const float* __restrict__ rec_hi,
                        float* __restrict__ out)
{
  __shared__ float bufA[N_A3BUF];   // holds a1 (2054), later a3 (8196)
  __shared__ float bufB[N_A2BUF];   // holds a2 (4102)

  const int row    = blockIdx.x;          // 0..1023 = (b,c)
  const int tid    = threadIdx.x;
  const int lane   = tid & 31;
  const int h      = lane >> 4;           // lane half (wave32)
  const int Ml     = lane & 15;
  // Wave index, forced into an SGPR so tile loops get scalar control flow.
  const int wave_s = __builtin_amdgcn_readfirstlane(tid >> 5);

  // Build the constant banded B matrices (one v2f per K-chunk per filter).
  v2f Ba[3], Bh[3];
#pragma unroll
  for (int c = 0; c < 3; ++c) {
    const int i0 = 4 * c + 2 * h;
    v2f ba, bh;
    ba.x = idwt_wgt(rec_lo, i0,     Ml);
    ba.y = idwt_wgt(rec_lo, i0 + 1, Ml);
    bh.x = idwt_wgt(rec_hi, i0,     Ml);
    bh.y = idwt_wgt(rec_hi, i0 + 1, Ml);
    Ba[c] = ba;  Bh[c] = bh;
  }

  // Warm the streaming inputs for this row into cache (global_prefetch_b8).
  __builtin_prefetch(d3 + (size_t)row * N_IN3, 0, 0);
  __builtin_prefetch(d2 + (size_t)row * N_IN2, 0, 0);

  // Level 1: approx(1030) + d0(1030) -> a1(2054) in bufA
  idwt_level<N_IN0>(approx + (size_t)row * N_IN0, d0 + (size_t)row * N_IN0,
                    bufA, tid, wave_s, h, Ml, Ba, Bh, rec_lo, rec_hi);
  __syncthreads();

  // Level 2: a1(2054) + d1(2054) -> a2(4102) in bufB
  idwt_level<N_IN1>(bufA, d1 + (size_t)row * N_IN1,
                    bufB, tid, wave_s, h, Ml, Ba, Bh, rec_lo, rec_hi);
  __syncthreads();

  // Level 3: a2 trimmed to 4101 + d2(4101) -> a3(8196) in bufA
  idwt_level<N_IN2>(bufB, d2 + (size_t)row * N_IN2,
                    bufA, tid, wave_s, h, Ml, Ba, Bh, rec_lo, rec_hi);
  __syncthreads();

  // Level 4: a3 trimmed to 8195 + d3(8195) -> out(16384), straight to HBM
  idwt_level<N_IN3>(bufA, d3 + (size_t)row * N_IN3,
                    out + (size_t)row * N_OUT, tid, wave_s, h, Ml,
                    Ba, Bh, rec_lo, rec_hi);
}

extern "C" void kernel_launch(void* const* d_in, const int* in_sizes, int n_in,
                              void* d_out, int out_size, void* d_ws, size_t ws_size,
                              hipStream_t stream) {
  const float* approx = (const float*)d_in[0];
  const float* d0     = (const float*)d_in[1];
  const float* d1     = (const float*)d_in[2];
  const float* d2     = (const float*)d_in[3];
  const float* d3     = (const float*)d_in[4];
  const float* rec_lo = (const float*)d_in[5];
  const float* rec_hi = (const float*)d_in[6];
  float* out = (float*)d_out;

  idwt4_fused_kernel<<<dim3(ROWS), dim3(256), 0, stream>>>(
      approx, d0, d1, d2, d3, rec_lo, rec_hi, out);
}